// LSTM_83511344103560
// MI455X (gfx1250) — compile-verified
//
#include <hip/hip_runtime.h>
#include <math.h>

typedef __attribute__((ext_vector_type(2))) float v2f;
typedef __attribute__((ext_vector_type(4))) float v4f;
typedef __attribute__((ext_vector_type(8))) float v8f;

#define Bsz   64
#define Tlen  512
#define Edim  256
#define Hdim  512
#define Odim  2
#define NWGL  32              // WGs per layer role (16 h-columns each)
#define NWGT  (2 * NWGL)      // total WGs in the persistent grid
#define NTHR  128             // 4 waves: one 16-row M tile each
#define BH    (Bsz * Hdim)    // 32768 floats per state buffer

#define NCX0  (Edim / 4)      // 64  K-chunks, layer0 input GEMM
#define NCH   (Hdim / 4)      // 128 K-chunks, all recurrent GEMMs
// LDS floats: WX tiles (4 gates * NC * 64) + WH tiles (4 * 128 * 64)
#define LDS_BYTES (4 * NCH * 64 * 2 * sizeof(float))  // 256KB (layer1 worst case)

__device__ __forceinline__ float sigf(float x) { return 1.0f / (1.0f + expf(-x)); }

__device__ __forceinline__ v8f wmma_f32(v2f a, v2f b, v8f c) {
  // D = A(16x4 f32) * B(4x16 f32) + C(16x16 f32)
  return __builtin_amdgcn_wmma_f32_16x16x4_f32(false, a, false, b, (short)0, c,
                                               false, false);
}

// Stage one weight slice (64 gate-rows x K) into LDS in WMMA-ready swizzled
// tile layout: tile (g, c) occupies 64 consecutive floats; lane L's v2f lives
// at tile*64 + L*2 and equals W[g*H + colbase + (L&15)][4c + (L>>4)*2 + {0,1}].
__device__ __forceinline__ void stage_weights(const float* __restrict__ W, int K,
                                              int NC, int colbase,
                                              float* __restrict__ ldsW) {
  for (int idx = threadIdx.x; idx < 64 * NC; idx += NTHR) {
    const int c = idx % NC;          // coalesced in c across lanes
    const int row = idx / NC;        // local gate-row 0..63
    const int g = row >> 4, n = row & 15;
    const float* src = W + (size_t)(g * Hdim + colbase + n) * K + c * 4;
    v4f w = *(const v4f*)src;
    float* tb = ldsW + (size_t)(g * NC + c) * 64;
    v2f lo; lo.x = w.x; lo.y = w.y;
    v2f hi; hi.x = w.z; hi.y = w.w;
    *(v2f*)(tb + n * 2) = lo;        // half 0 (k%4 in {0,1})
    *(v2f*)(tb + 32 + n * 2) = hi;   // half 1 (k%4 in {2,3})
  }
}

// K-chain with B from LDS and a DEPTH-deep register prefetch pipeline on the
// global A stream. lb must already include +lane*2.
template <int NC>
__device__ __forceinline__ void gemm4_lds(const float* __restrict__ ag,
                                          const float* __restrict__ lb,
                                          v8f* __restrict__ acc) {
  constexpr int DEPTH = 16;  // NC is 64 or 128: both multiples of 2*DEPTH
#pragma unroll
  for (int d = 0; d < DEPTH; ++d) {
    // abuf lives in registers via full unrolling below
  }
  v2f abuf[DEPTH];
#pragma unroll
  for (int d = 0; d < DEPTH; ++d) abuf[d] = *(const v2f*)(ag + d * 4);

  for (int c0 = 0; c0 < NC - DEPTH; c0 += DEPTH) {
#pragma unroll
    for (int i = 0; i < DEPTH; ++i) {
      const int c = c0 + i;
      v2f an = *(const v2f*)(ag + (c + DEPTH) * 4);
      v2f b0 = *(const v2f*)(lb + (0 * NC + c) * 64);
      v2f b1 = *(const v2f*)(lb + (1 * NC + c) * 64);
      v2f b2 = *(const v2f*)(lb + (2 * NC + c) * 64);
      v2f b3 = *(const v2f*)(lb + (3 * NC + c) * 64);
      acc[0] = wmma_f32(abuf[i], b0, acc[0]);
      acc[1] = wmma_f32(abuf[i], b1, acc[1]);
      acc[2] = wmma_f32(abuf[i], b2, acc[2]);
      acc[3] = wmma_f32(abuf[i], b3, acc[3]);
      abuf[i] = an;
    }
  }
#pragma unroll
  for (int i = 0; i < DEPTH; ++i) {
    const int c = NC - DEPTH + i;
    v2f b0 = *(const v2f*)(lb + (0 * NC + c) * 64);
    v2f b1 = *(const v2f*)(lb + (1 * NC + c) * 64);
    v2f b2 = *(const v2f*)(lb + (2 * NC + c) * 64);
    v2f b3 = *(const v2f*)(lb + (3 * NC + c) * 64);
    acc[0] = wmma_f32(abuf[i], b0, acc[0]);
    acc[1] = wmma_f32(abuf[i], b1, acc[1]);
    acc[2] = wmma_f32(abuf[i], b2, acc[2]);
    acc[3] = wmma_f32(abuf[i], b3, acc[3]);
  }
}

// Sense-reversing grid-wide barrier across the persistent grid.
__device__ __forceinline__ void global_barrier(int* cnt, int* gen) {
  __syncthreads();
  if (threadIdx.x == 0) {
    __threadfence();
    int g = __hip_atomic_load(gen, __ATOMIC_ACQUIRE, __HIP_MEMORY_SCOPE_AGENT);
    int prev = __hip_atomic_fetch_add(cnt, 1, __ATOMIC_ACQ_REL, __HIP_MEMORY_SCOPE_AGENT);
    if (prev == NWGT - 1) {
      __hip_atomic_store(cnt, 0, __ATOMIC_RELAXED, __HIP_MEMORY_SCOPE_AGENT);
      __hip_atomic_fetch_add(gen, 1, __ATOMIC_ACQ_REL, __HIP_MEMORY_SCOPE_AGENT);
    } else {
      while (__hip_atomic_load(gen, __ATOMIC_ACQUIRE, __HIP_MEMORY_SCOPE_AGENT) == g) {
        __builtin_amdgcn_s_sleep(1);
      }
    }
    __threadfence();
  }
  __syncthreads();
}

__global__ void __launch_bounds__(256)
lstm_init_ws(float* ws) {
  size_t n = 6u * BH;  // h0[2], h1[2], c0, c1
  for (size_t i = (size_t)blockIdx.x * blockDim.x + threadIdx.x; i < n;
       i += (size_t)gridDim.x * blockDim.x)
    ws[i] = 0.0f;
  if (blockIdx.x == 0 && threadIdx.x == 0) {
    int* bar = (int*)(ws + 6u * BH);
    bar[0] = 0;  // count
    bar[1] = 0;  // generation
  }
}

__global__ void __launch_bounds__(NTHR)
lstm_persistent(const int* __restrict__ x, const float* __restrict__ emb,
                const float* __restrict__ Wih0, const float* __restrict__ Whh0,
                const float* __restrict__ b0,
                const float* __restrict__ Wih1, const float* __restrict__ Whh1,
                const float* __restrict__ b1,
                const float* __restrict__ Wclf, const float* __restrict__ bclf,
                float* __restrict__ out, float* __restrict__ ws) {
  extern __shared__ float smem[];

  const int lane = threadIdx.x & 31;
  const int mt   = threadIdx.x >> 5;   // 4 waves: one M tile each
  const bool isL1 = blockIdx.x >= NWGL;
  const int wg   = blockIdx.x & (NWGL - 1);

  const int am = lane & 15;            // A row within tile
  const int ak = (lane >> 4) << 1;     // A k-pair offset within K=4 chunk
  const int cn = lane & 15;            // C/D col
  const int chalf = (lane >> 4) * 8;   // C/D row half select

  const int colbase = wg * 16;               // owned h column base
  const int arow = mt * 16 + am;             // batch row this lane feeds into A
  const int crow0 = mt * 16 + chalf;         // accum row base (+v)
  const int col = colbase + cn;

  float* h0buf = ws;                 // 2*BH
  float* h1buf = ws + 2u * BH;       // 2*BH
  float* c0    = ws + 4u * BH;       // BH
  float* c1    = ws + 5u * BH;       // BH
  int*   bar   = (int*)(ws + 6u * BH);

  // Role-dependent weights: layer0 = (Wih0 ld E, Whh0), layer1 = (Wih1, Whh1)
  const float* WX = isL1 ? Wih1 : Wih0;
  const float* WH = isL1 ? Whh1 : Whh0;
  const float* BB = isL1 ? b1 : b0;
  const int    ldx = isL1 ? Hdim : Edim;
  const int    ncx = isL1 ? NCH : NCX0;
  float* cbuf = isL1 ? c1 : c0;

  // Stage this WG's weight slices into LDS (once for the whole scan)
  float* ldsWX = smem;                       // 4*ncx*64 floats
  float* ldsWH = smem + (size_t)4 * ncx * 64;  // 4*128*64 floats
  stage_weights(WX, ldx, ncx, colbase, ldsWX);
  stage_weights(WH, Hdim, NCH, colbase, ldsWH);
  __syncthreads();

  const float* lbx = ldsWX + lane * 2;
  const float* lbh = ldsWH + lane * 2;

  float bias[4];
#pragma unroll
  for (int g = 0; g < 4; ++g) bias[g] = BB[g * Hdim + colbase + cn];

  // Phase p: layer0 computes step t=p (p<T); layer1 computes step t=p-1 (p>=1).
  for (int p = 0; p <= Tlen; ++p) {
    const int r = p & 1;
    const float* h0r = h0buf + (size_t)r * BH;
    float*       h0w = h0buf + (size_t)(r ^ 1) * BH;
    const float* h1r = h1buf + (size_t)r * BH;
    float*       h1w = h1buf + (size_t)(r ^ 1) * BH;

    if (!isL1) {
      if (p < Tlen) {
        v8f acc[4];
#pragma unroll
        for (int g = 0; g < 4; ++g)
#pragma unroll
          for (int v = 0; v < 8; ++v) acc[g][v] = bias[g];

        const int tok = x[arow * Tlen + p];
        gemm4_lds<NCX0>(emb + (size_t)tok * Edim + ak, lbx, acc);
        gemm4_lds<NCH>(h0r + (size_t)arow * Hdim + ak, lbh, acc);
#pragma unroll
        for (int v = 0; v < 8; ++v) {
          const size_t idx = (size_t)(crow0 + v) * Hdim + col;
          float iv = sigf(acc[0][v]);
          float fv = sigf(acc[1][v]);
          float gv = tanhf(acc[2][v]);
          float ov = sigf(acc[3][v]);
          float cc = fv * cbuf[idx] + iv * gv;
          cbuf[idx] = cc;
          h0w[idx] = ov * tanhf(cc);
        }
      }
    } else {
      if (p >= 1) {
        v8f acc[4];
#pragma unroll
        for (int g = 0; g < 4; ++g)
#pragma unroll
          for (int v = 0; v < 8; ++v) acc[g][v] = bias[g];

        // A = h0(t) (written last phase), then h1(t-1)
        gemm4_lds<NCH>(h0r + (size_t)arow * Hdim + ak, lbx, acc);
        gemm4_lds<NCH>(h1r + (size_t)arow * Hdim + ak, lbh, acc);
#pragma unroll
        for (int v = 0; v < 8; ++v) {
          const size_t idx = (size_t)(crow0 + v) * Hdim + col;
          float iv = sigf(acc[0][v]);
          float fv = sigf(acc[1][v]);
          float gv = tanhf(acc[2][v]);
          float ov = sigf(acc[3][v]);
          float cc = fv * cbuf[idx] + iv * gv;
          cbuf[idx] = cc;
          h1w[idx] = ov * tanhf(cc);
        }
      }
    }

    global_barrier(bar, bar + 1);  // one barrier per pipeline phase
  }

  // Classifier on last h1 (phase T=512 wrote buffer (0^1)=1). Tiny: 64x2.
  if (blockIdx.x == NWGL) {
    const float* hl = h1buf + (size_t)BH;
    for (int i = threadIdx.x; i < Bsz * Odim; i += blockDim.x) {
      const int b = i / Odim, o = i % Odim;
      float s = bclf[o];
      const float* hr = hl + (size_t)b * Hdim;
      const float* wr = Wclf + (size_t)o * Hdim;
      for (int k = 0; k < Hdim; ++k) s += hr[k] * wr[k];
      out[i] = s;
    }
  }
}

extern "C" void kernel_launch(void* const* d_in, const int* in_sizes, int n_in,
                              void* d_out, int out_size, void* d_ws, size_t ws_size,
                              hipStream_t stream) {
  (void)in_sizes; (void)n_in; (void)out_size; (void)ws_size;
  const int*   x    = (const int*)d_in[0];
  const float* emb  = (const float*)d_in[1];
  const float* Wih0 = (const float*)d_in[2];
  const float* Whh0 = (const float*)d_in[3];
  const float* b0   = (const float*)d_in[4];
  const float* Wih1 = (const float*)d_in[5];
  const float* Whh1 = (const float*)d_in[6];
  const float* b1   = (const float*)d_in[7];
  const float* Wclf = (const float*)d_in[8];
  const float* bclf = (const float*)d_in[9];
  float* out = (float*)d_out;
  float* ws  = (float*)d_ws;

  // Allow the 256KB dynamic-LDS allocation (CDNA5 WGP has 320KB).
  (void)hipFuncSetAttribute((const void*)lstm_persistent,
                            hipFuncAttributeMaxDynamicSharedMemorySize,
                            (int)LDS_BYTES);

  lstm_init_ws<<<64, 256, 0, stream>>>(ws);
  lstm_persistent<<<NWGT, NTHR, LDS_BYTES, stream>>>(
      x, emb, Wih0, Whh0, b0, Wih1, Whh1, b1, Wclf, bclf, out, ws);
}